// MultiHeadAttentionQuantum_65481071403185
// MI455X (gfx1250) — compile-verified
//
#include <hip/hip_runtime.h>
#include <hip/hip_bf16.h>

typedef float v2f __attribute__((ext_vector_type(2)));
typedef float v8f __attribute__((ext_vector_type(8)));

#define EDIM 128          // embedding = GEMM K = GEMM N
#define MROWS 4096        // B*S
#define KSTEPS 32         // 128 / 4 (K per V_WMMA_F32_16X16X4_F32)

// Fused: z = cos(qp[e%8,1]) * cos(x + qp[e%8,0])  (closed-form product-state
// circuit), then out = z @ w_out^T + b_out via fp32 WMMA 16x16x4.
__global__ __launch_bounds__(256)
void qattn_fused_wmma(const float* __restrict__ x,
                      const float* __restrict__ qp,     // [8,3]
                      const float* __restrict__ w,      // [128,128] row-major
                      const float* __restrict__ bvec,   // [128]
                      float* __restrict__ out)          // [4096,128]
{
    const int lane = threadIdx.x & 31;
    const int wave = threadIdx.x >> 5;
    const int mt   = blockIdx.x * 8 + wave;   // 0..255 row tiles
    const int m0   = mt * 16;
    const int h    = lane >> 4;               // K-half select (ISA A/B layout)
    const int l15  = lane & 15;
    const int mrow = m0 + l15;                // A-matrix row for this lane

    // Gate constants. Element k uses qp[k&7]; here k = 4*s + 2*h + j, so
    // (k&7) = 4*(s&1) + 2*h + j  -> only s-parity varies at runtime.
    float qa[2][2], cb[2][2];
#pragma unroll
    for (int p = 0; p < 2; ++p) {
#pragma unroll
        for (int j = 0; j < 2; ++j) {
            const int kk = 4 * p + 2 * h + j;
            qa[p][j] = qp[kk * 3 + 0];
            cb[p][j] = cosf(qp[kk * 3 + 1]);
        }
    }

    // Build all A fragments for K=0..127 once (reused across 8 N-tiles).
    // A layout (16x4 f32): lanes 0-15 hold K=4s+{0,1}, lanes 16-31 K=4s+{2,3}.
    v2f a[KSTEPS];
    const float* xrow = x + (size_t)mrow * EDIM + 2 * h;
#pragma unroll
    for (int s = 0; s < KSTEPS; ++s) {
        const v2f xv = *(const v2f*)(xrow + 4 * s);
        const int p = s & 1;
        v2f t;
        t.x = cb[p][0] * cosf(xv.x + qa[p][0]);
        t.y = cb[p][1] * cosf(xv.y + qa[p][1]);
        a[s] = t;
    }

    // Sweep the 8 column tiles of the output.
    for (int nt = 0; nt < 8; ++nt) {
        const int n = nt * 16 + l15;
        // B layout (4x16 f32): VGPR j, lane-half h -> row K = 4s + j + 2h.
        // B[k,n] = w_out[n,k], so per lane a contiguous float2 of w row n.
        const float* wrow = w + (size_t)n * EDIM + 2 * h;
        v8f acc = {};
#pragma unroll
        for (int s = 0; s < KSTEPS; ++s) {
            const v2f bv = *(const v2f*)(wrow + 4 * s);
            acc = __builtin_amdgcn_wmma_f32_16x16x4_f32(
                false, a[s], false, bv, (short)0, acc, false, false);
        }
        // D layout: VGPR r -> row m0 + r + 8h, col = n (lane&15). Coalesced.
        const float bias = bvec[n];
#pragma unroll
        for (int r = 0; r < 8; ++r) {
            const int m = m0 + r + 8 * h;
            out[(size_t)m * EDIM + n] = acc[r] + bias;
        }
    }
}

extern "C" void kernel_launch(void* const* d_in, const int* in_sizes, int n_in,
                              void* d_out, int out_size, void* d_ws, size_t ws_size,
                              hipStream_t stream) {
    (void)in_sizes; (void)n_in; (void)out_size; (void)d_ws; (void)ws_size;
    const float* x    = (const float*)d_in[0];   // [4,1024,128]
    const float* qp   = (const float*)d_in[1];   // [8,3]
    const float* wout = (const float*)d_in[2];   // [128,128]
    const float* bout = (const float*)d_in[3];   // [128]
    float* out = (float*)d_out;                  // [4,1024,128]

    // 256 row-tiles of 16 rows; 8 waves (wave32) per 256-thread block.
    qattn_fused_wmma<<<32, 256, 0, stream>>>(x, qp, wout, bout, out);
}